// MultiModalPrototypicalLoss_86208583566041
// MI455X (gfx1250) — compile-verified
//
#include <hip/hip_runtime.h>
#include <math.h>

typedef __attribute__((ext_vector_type(2))) float v2f;
typedef __attribute__((ext_vector_type(8))) float v8f;

#define NROW   4096          // N = 2*B
#define HALFB  2048
#define DIM    128
#define NCLS   512
#define PLANE  (NROW * DIM)
#define NSTRIP 16            // column strips in gram kernel (256 cols each)
#define W0 0.4f
#define W1 0.3f
#define W2 0.3f

__device__ __forceinline__ float wave_sum32(float v) {
#pragma unroll
    for (int m = 1; m < 32; m <<= 1) v += __shfl_xor(v, m, 32);
    return v;
}
// sum within each 16-lane half (cols of a WMMA C tile)
__device__ __forceinline__ float half_sum16(float v) {
#pragma unroll
    for (int m = 1; m < 16; m <<= 1) v += __shfl_xor(v, m, 32);
    return v;
}
__device__ __forceinline__ float wave_max32(float v) {
#pragma unroll
    for (int m = 1; m < 32; m <<= 1) v = fmaxf(v, __shfl_xor(v, m, 32));
    return v;
}

// ---------------- K1: l2-normalize + pack feats [3][4096][128] ----------------
__global__ void k_normalize(const float* __restrict__ s_i, const float* __restrict__ s_j,
                            const float* __restrict__ c_i, const float* __restrict__ c_j,
                            const float* __restrict__ p_i, const float* __restrict__ p_j,
                            float* __restrict__ fnorm) {
    const int wave = (blockIdx.x * blockDim.x + threadIdx.x) >> 5;  // 0 .. 3*4096-1
    const int lane = threadIdx.x & 31;
    const int m = wave / NROW;
    const int r = wave % NROW;
    const float* src;
    if (m == 0)      src = (r < HALFB) ? s_i + (size_t)r * DIM : s_j + (size_t)(r - HALFB) * DIM;
    else if (m == 1) src = (r < HALFB) ? c_i + (size_t)r * DIM : c_j + (size_t)(r - HALFB) * DIM;
    else             src = (r < HALFB) ? p_i + (size_t)r * DIM : p_j + (size_t)(r - HALFB) * DIM;
    float x0 = src[lane * 4 + 0], x1 = src[lane * 4 + 1];
    float x2 = src[lane * 4 + 2], x3 = src[lane * 4 + 3];
    float ss = wave_sum32(x0 * x0 + x1 * x1 + x2 * x2 + x3 * x3);
    float inv = 1.0f / fmaxf(sqrtf(ss), 1e-12f);
    float* dst = fnorm + (size_t)m * PLANE + (size_t)r * DIM + lane * 4;
    dst[0] = x0 * inv; dst[1] = x1 * inv; dst[2] = x2 * inv; dst[3] = x3 * inv;
}

// load one k-chunk: 3 A operands (one per modality) + 4 tiles x 3 modalities B operands
#define LOADC(A_, B_, kk)                                                         \
    {                                                                             \
        _Pragma("unroll") for (int m_ = 0; m_ < 3; ++m_) {                        \
            (A_)[m_] = *(const v2f*)(Abase + (kk) + m_ * PLANE);                  \
            _Pragma("unroll") for (int t_ = 0; t_ < 4; ++t_)                      \
                (B_)[t_][m_] = *(const v2f*)(Bbase + t_ * 16 * DIM + (kk) + m_ * PLANE); \
        }                                                                         \
    }

#define WMMAC(A_, B_)                                                             \
    {                                                                             \
        _Pragma("unroll") for (int m_ = 0; m_ < 3; ++m_)                          \
            _Pragma("unroll") for (int t_ = 0; t_ < 4; ++t_)                      \
                acc[m_][t_] = __builtin_amdgcn_wmma_f32_16x16x4_f32(              \
                    false, (A_)[m_], false, (B_)[t_][m_], (short)0,               \
                    acc[m_][t_], false, false);                                   \
    }

// ---------------- K2: fused 3x Gram -> wsim -> exp -> row partial ----------------
// one wave per (16-row slab, 256-col strip); strip = 4 groups of 4 column tiles.
// Per group: 12 independent WMMA accumulators, 2-deep software-pipelined K loop.
__global__ void __launch_bounds__(32, 1)
k_gram(const float* __restrict__ fnorm, float* __restrict__ part1) {
    const int lane  = threadIdx.x;               // 0..31, full wave
    const int wid   = blockIdx.x;                // 256 slabs * 16 strips
    const int slab  = wid / NSTRIP;
    const int strip = wid % NSTRIP;
    const int m0    = slab * 16;
    const int kOff  = (lane >> 4) * 2;           // lanes 16-31 hold K+2,K+3
    const float* Abase = fnorm + (size_t)(m0 + (lane & 15)) * DIM + kOff;

    float racc[8] = {0.f, 0.f, 0.f, 0.f, 0.f, 0.f, 0.f, 0.f};

    for (int grp = 0; grp < 4; ++grp) {
        const int n0 = strip * 256 + grp * 64;   // 4 tiles => 64 columns
        const float* Bbase = fnorm + (size_t)(n0 + (lane & 15)) * DIM + kOff;

        v8f acc[3][4];
#pragma unroll
        for (int m = 0; m < 3; ++m)
#pragma unroll
            for (int t = 0; t < 4; ++t) acc[m][t] = (v8f){};

        v2f aC[3], bC[4][3], aN[3], bN[4][3];
        LOADC(aC, bC, 0);
#pragma unroll 1
        for (int k = 0; k < DIM - 8; k += 8) {
            LOADC(aN, bN, k + 4);    // prefetch chunk k+4
            WMMAC(aC, bC);           // consume chunk k
            LOADC(aC, bC, k + 8);    // prefetch chunk k+8
            WMMAC(aN, bN);           // consume chunk k+4
        }
        LOADC(aN, bN, DIM - 4);
        WMMAC(aC, bC);               // chunk DIM-8
        WMMAC(aN, bN);               // chunk DIM-4

#pragma unroll
        for (int t = 0; t < 4; ++t) {
#pragma unroll
            for (int v = 0; v < 8; ++v) {
                float s0 = acc[0][t][v], s1 = acc[1][t][v], s2 = acc[2][t][v];
                float wn = 0.f, wd = 0.f;
                if (s0 != 0.f) { wn += W0 * s0; wd += W0; }
                if (s1 != 0.f) { wn += W1 * s1; wd += W1; }
                if (s2 != 0.f) { wn += W2 * s2; wd += W2; }
                racc[v] += expf(2.0f * (wn / (wd + 1e-12f)));   // /TEMP(0.5) == *2
            }
        }
    }
    // C layout: vgpr v holds row (m0 + v + 8*(lane>=16)), col = lane&15
#pragma unroll
    for (int v = 0; v < 8; ++v) racc[v] = half_sum16(racc[v]);
    if ((lane & 15) == 0) {
        const int rbase = m0 + ((lane >> 4) ? 8 : 0);
#pragma unroll
        for (int v = 0; v < 8; ++v)
            part1[(size_t)(rbase + v) * NSTRIP + strip] = racc[v];
    }
}

// proto loads: B per modality comes from a different centroid array
#define LOADP(A_, B_, kk)                                                         \
    {                                                                             \
        _Pragma("unroll") for (int m_ = 0; m_ < 3; ++m_) {                        \
            (A_)[m_] = *(const v2f*)(Abase + (kk) + m_ * PLANE);                  \
            _Pragma("unroll") for (int t_ = 0; t_ < 4; ++t_)                      \
                (B_)[t_][m_] = *(const v2f*)(cb[m_] + t_ * 16 * DIM + (kk));      \
        }                                                                         \
    }

// ---------------- K3: proto logits lw[4096][512] via WMMA vs centroids ----------------
// one wave per (16-row slab, 64-class group): 256 * 8 = 2048 waves
__global__ void __launch_bounds__(32, 1)
k_proto(const float* __restrict__ fnorm,
        const float* __restrict__ cen_st,
        const float* __restrict__ cen_car,
        const float* __restrict__ cen_pl,
        const float* __restrict__ conc,
        float* __restrict__ lw) {
    const int lane = threadIdx.x;
    const int wid  = blockIdx.x;                 // 256 slabs * 8 groups
    const int slab = wid >> 3;
    const int grp  = wid & 7;
    const int m0   = slab * 16;
    const int n0   = grp * 64;                   // 4 class tiles
    const int kOff = (lane >> 4) * 2;
    const float* Abase = fnorm + (size_t)(m0 + (lane & 15)) * DIM + kOff;
    const size_t boff  = (size_t)(n0 + (lane & 15)) * DIM + kOff;
    const float* cb[3] = {cen_st + boff, cen_car + boff, cen_pl + boff};

    v8f acc[3][4];
#pragma unroll
    for (int m = 0; m < 3; ++m)
#pragma unroll
        for (int t = 0; t < 4; ++t) acc[m][t] = (v8f){};

    v2f aC[3], bC[4][3], aN[3], bN[4][3];
    LOADP(aC, bC, 0);
#pragma unroll 1
    for (int k = 0; k < DIM - 8; k += 8) {
        LOADP(aN, bN, k + 4);
        WMMAC(aC, bC);
        LOADP(aC, bC, k + 8);
        WMMAC(aN, bN);
    }
    LOADP(aN, bN, DIM - 4);
    WMMAC(aC, bC);
    WMMAC(aN, bN);

    const int rowbase = m0 + ((lane >> 4) ? 8 : 0);
#pragma unroll
    for (int t = 0; t < 4; ++t) {
        const int cls = n0 + t * 16 + (lane & 15);
        const float t0 = conc[cls];
        const float t1 = conc[NCLS + cls];
        const float t2 = conc[2 * NCLS + cls];
#pragma unroll
        for (int v = 0; v < 8; ++v) {
            float wn = 0.f, wd = 0.f;
            float l0 = acc[0][t][v] / (t0 + 1e-12f);
            if (!((t0 == 0.f) || (l0 == 0.f))) { wn += W0 * l0; wd += W0; }
            float l1 = acc[1][t][v] / (t1 + 1e-12f);
            if (!((t1 == 0.f) || (l1 == 0.f))) { wn += W1 * l1; wd += W1; }
            float l2 = acc[2][t][v] / (t2 + 1e-12f);
            if (!((t2 == 0.f) || (l2 == 0.f))) { wn += W2 * l2; wd += W2; }
            lw[(size_t)(rowbase + v) * NCLS + cls] = wn / (wd + 1e-12f);
        }
    }
}

// ---------------- K4: per-row finalize: rowsum, diag/pos dots, proto logsumexp ----------------
__global__ void k_rowfin(const float* __restrict__ fnorm,
                         const float* __restrict__ part1,
                         const float* __restrict__ lw,
                         const int* __restrict__ labels,
                         float* __restrict__ lossrow) {
    const int i    = blockIdx.x;
    const int lane = threadIdx.x;
    // deterministic rowsum of exp-sims
    float p = (lane < NSTRIP) ? part1[(size_t)i * NSTRIP + lane] : 0.f;
    p = wave_sum32(p);
    // diag & positive-pair dots, per modality
    const int j = (i + HALFB) & (NROW - 1);
    float sself[3], spos[3];
#pragma unroll
    for (int m = 0; m < 3; ++m) {
        const float* fi = fnorm + (size_t)m * PLANE + (size_t)i * DIM + lane * 4;
        const float* fj = fnorm + (size_t)m * PLANE + (size_t)j * DIM + lane * 4;
        float ds = 0.f, dp = 0.f;
#pragma unroll
        for (int t = 0; t < 4; ++t) { float a = fi[t]; ds += a * a; dp += a * fj[t]; }
        sself[m] = wave_sum32(ds);
        spos[m]  = wave_sum32(dp);
    }
    // proto: logsumexp over all 512 classes (proto_ids is a permutation)
    const float* lrow = lw + (size_t)i * NCLS;
    float mx = -3.402823466e38f;
    for (int c = lane; c < NCLS; c += 32) mx = fmaxf(mx, lrow[c]);
    mx = wave_max32(mx);
    float se = 0.f;
    for (int c = lane; c < NCLS; c += 32) se += expf(lrow[c] - mx);
    se = wave_sum32(se);

    if (lane == 0) {
        float wnD = 0.f, wdD = 0.f, wnP = 0.f, wdP = 0.f;
        if (sself[0] != 0.f) { wnD += W0 * sself[0]; wdD += W0; }
        if (sself[1] != 0.f) { wnD += W1 * sself[1]; wdD += W1; }
        if (sself[2] != 0.f) { wnD += W2 * sself[2]; wdD += W2; }
        if (spos[0]  != 0.f) { wnP += W0 * spos[0];  wdP += W0; }
        if (spos[1]  != 0.f) { wnP += W1 * spos[1];  wdP += W1; }
        if (spos[2]  != 0.f) { wnP += W2 * spos[2];  wdP += W2; }
        float eD = expf(2.0f * (wnD / (wdD + 1e-12f)));
        float eP = expf(2.0f * (wnP / (wdP + 1e-12f)));
        // -log(pos/(pos+neg_sum)) with pos+neg_sum = rowsum - diag
        float loss1 = logf(p - eD) - logf(eP);
        int lab = labels[i];
        float loss2 = mx + logf(se) - lrow[lab];
        lossrow[i] = loss1 + loss2;
    }
}

// ---------------- K5: deterministic final mean ----------------
__global__ void k_final(const float* __restrict__ lossrow, float* __restrict__ out) {
    __shared__ float sh[256];
    float s = 0.f;
    for (int i = threadIdx.x; i < NROW; i += 256) s += lossrow[i];
    sh[threadIdx.x] = s;
    __syncthreads();
    for (int step = 128; step > 0; step >>= 1) {
        if ((int)threadIdx.x < step) sh[threadIdx.x] += sh[threadIdx.x + step];
        __syncthreads();
    }
    if (threadIdx.x == 0) out[0] = sh[0] / (float)NROW;
}

extern "C" void kernel_launch(void* const* d_in, const int* in_sizes, int n_in,
                              void* d_out, int out_size, void* d_ws, size_t ws_size,
                              hipStream_t stream) {
    const float* st_i = (const float*)d_in[0];
    const float* st_j = (const float*)d_in[1];
    const float* ca_i = (const float*)d_in[2];
    const float* ca_j = (const float*)d_in[3];
    const float* pl_i = (const float*)d_in[4];
    const float* pl_j = (const float*)d_in[5];
    const int*   lab  = (const int*)d_in[6];
    const float* c_st = (const float*)d_in[7];
    const float* c_ca = (const float*)d_in[8];
    const float* c_pl = (const float*)d_in[9];
    const float* conc = (const float*)d_in[10];

    float* fnorm   = (float*)d_ws;                    // 3 * 4096 * 128
    float* part1   = fnorm + (size_t)3 * PLANE;       // 4096 * 16
    float* lw      = part1 + (size_t)NROW * NSTRIP;   // 4096 * 512
    float* lossrow = lw + (size_t)NROW * NCLS;        // 4096

    k_normalize<<<(3 * NROW) / 8, 256, 0, stream>>>(st_i, st_j, ca_i, ca_j, pl_i, pl_j, fnorm);
    k_gram<<<(NROW / 16) * NSTRIP, 32, 0, stream>>>(fnorm, part1);
    k_proto<<<(NROW / 16) * (NCLS / 64), 32, 0, stream>>>(fnorm, c_st, c_ca, c_pl, conc, lw);
    k_rowfin<<<NROW, 32, 0, stream>>>(fnorm, part1, lw, lab, lossrow);
    k_final<<<1, 256, 0, stream>>>(lossrow, (float*)d_out);
}